// MoveNetLoss_41051297415596
// MI455X (gfx1250) — compile-verified
//
#include <hip/hip_runtime.h>
#include <math.h>

// MoveNet-style multi-term loss, fused for MI455X (gfx1250, wave32).
// Bone loss computed via V_WMMA_F32_16X16X4_F32: D = M(16x17 limb incidence) x E(17x16 pixels).

typedef __attribute__((ext_vector_type(2))) float v2f;
typedef __attribute__((ext_vector_type(8))) float v8f;

#define HW     2304        // 48*48
#define WIDTH  48
#define NBATCH 128
#define NB2    2304        // blocks in fused kernel (18432 waves * 16 px = 294912 px)

__constant__ int c_LA[16] = {5,5,7,6,8,11,5,6,11,13,12,14,0,0,1,2};
__constant__ int c_LB[16] = {6,7,9,8,10,12,11,12,13,15,14,16,1,2,3,4};

// Focal term accumulated into s; returns e = sigmoid(l) - t (needed for bone loss).
__device__ __forceinline__ float focal_e(float l, float t, float& s) {
  float emb = expf(-fabsf(l));                 // e^{-|l|}
  float bce = fmaxf(l, 0.f) - l * t + log1pf(emb);
  float pl  = 1.f / (1.f + emb);               // sigmoid(|l|)
  float p   = (l >= 0.f) ? pl : 1.f - pl;      // sigmoid(l)
  float pt  = fminf(fmaxf(p * t + (1.f - p) * (1.f - t), 1e-6f), 1.f - 1e-6f);
  float at  = 0.25f * t + 0.75f * (1.f - t);
  float om  = 1.f - pt;
  s += at * om * om * bce;
  return p - t;
}

// ---------------- Kernel 1: per-batch argmax over gt_ct (labels ch 17) --------------
__global__ __launch_bounds__(256) void k_argmax(const float* __restrict__ lab,
                                                int* __restrict__ cyx) {
  __shared__ float sv[256];
  __shared__ int   si[256];
  const int b = blockIdx.x, tid = threadIdx.x;
  const float* g = lab + (size_t)b * 86 * HW + (size_t)17 * HW;
  float best = -INFINITY; int bi = HW;
  for (int i = tid; i < HW; i += 256) {
    float v = g[i];
    if (v > best) { best = v; bi = i; }          // strided ascending -> first max kept
  }
  sv[tid] = best; si[tid] = bi; __syncthreads();
  for (int s = 128; s > 0; s >>= 1) {
    if (tid < s) {
      float v2 = sv[tid + s]; int i2 = si[tid + s];
      if (v2 > sv[tid] || (v2 == sv[tid] && i2 < si[tid])) { sv[tid] = v2; si[tid] = i2; }
    }
    __syncthreads();
  }
  if (tid == 0) { cyx[2 * b] = si[0] / WIDTH; cyx[2 * b + 1] = si[0] % WIDTH; }
}

// ------- Kernel 2: fused focal(hm), focal(ct), bone loss via f32 WMMA ---------------
// Each wave handles 16 consecutive pixels. Lane L: pixel n = L&15.
// WMMA f32 16x16x4 operand layout (ISA 7.12.2):
//   A 16x4: lanes 0-15 hold M=0..15, {VGPR0,VGPR1} = K{0,1}; lanes 16-31 = K{2,3}
//   B 4x16: lanes 0-15 hold N=0..15, {VGPR0,VGPR1} = K{0,1}; lanes 16-31 = K{2,3}
// So per K-chunk kk, lane L supplies channels c = kk*4 + (L>=16?2:0) + {0,1},
// which is exactly the channel split used for the focal sums (full disjoint cover of 0..16).
__global__ __launch_bounds__(256) void k_focal_bone(const float* __restrict__ hm,
                                                    const float* __restrict__ ct,
                                                    const float* __restrict__ lab,
                                                    float* __restrict__ p_hm,
                                                    float* __restrict__ p_ct,
                                                    float* __restrict__ p_bone) {
  const int tid  = threadIdx.x;
  const int lane = tid & 31;
  const int wave = tid >> 5;
  const int n    = lane & 15;          // pixel-in-group / matrix row / matrix col
  const int hi   = lane >> 4;          // which K half this lane supplies
  const int group = blockIdx.x * 8 + wave;       // 0..18431
  const int pix16 = group * 16;                  // 16 | 2304 -> never crosses batch
  const int b   = pix16 / HW;
  const int pix = (pix16 - b * HW) + n;

  const float* hmb  = hm  + (size_t)b * 17 * HW + pix;
  const float* labb = lab + (size_t)b * 86 * HW + pix;
  const int la = c_LA[n], lb = c_LB[n];          // A-matrix row m == n

  float s_hm = 0.f, s_ct = 0.f, s_bone = 0.f;
  v8f acc = {};
#pragma unroll
  for (int kk = 0; kk < 5; ++kk) {               // K = 17 padded to 20
    const int cb = kk * 4 + hi * 2;
    float e0 = 0.f, e1 = 0.f;
    if (cb < 17) e0 = focal_e(hmb[(size_t)cb * HW],       labb[(size_t)cb * HW],       s_hm);
    if (cb + 1 < 17) e1 = focal_e(hmb[(size_t)(cb+1)*HW], labb[(size_t)(cb+1)*HW],     s_hm);
    v2f av = { (float)((cb     == la) - (cb     == lb)),
               (float)((cb + 1 == la) - (cb + 1 == lb)) };   // limb incidence, 0 past K=16
    v2f bv = { e0, e1 };
    acc = __builtin_amdgcn_wmma_f32_16x16x4_f32(false, av, false, bv,
                                                (short)0, acc, false, false);
  }
  // center focal: one lane per pixel (lower half)
  if (hi == 0) {
    (void)focal_e(ct[(size_t)b * HW + pix], labb[(size_t)17 * HW], s_ct);
  }
  // sum of squares of D = sum over (limb, pixel) of ((dp - dg)^2)
#pragma unroll
  for (int i = 0; i < 8; ++i) s_bone += acc[i] * acc[i];

  __shared__ float red[256];
  auto reduce_store = [&](float v, float* dst) {
    red[tid] = v; __syncthreads();
    for (int s = 128; s > 0; s >>= 1) { if (tid < s) red[tid] += red[tid + s]; __syncthreads(); }
    if (tid == 0) dst[blockIdx.x] = red[0];
    __syncthreads();
  };
  reduce_store(s_hm,   p_hm);
  reduce_store(s_ct,   p_ct);
  reduce_store(s_bone, p_bone);
}

// ---------------- Kernel 3: reg + off gather losses (tiny) --------------------------
__global__ __launch_bounds__(256) void k_regoff(const float* __restrict__ regs,
                                                const float* __restrict__ offs,
                                                const float* __restrict__ lab,
                                                const int* __restrict__ masks,
                                                const int* __restrict__ cyx,
                                                float* __restrict__ s3) {
  const int tid = threadIdx.x;
  float reg_num = 0.f, off_num = 0.f, msum = 0.f;
  for (int p = tid; p < NBATCH * 17; p += 256) {
    const int b = p / 17, k = p - b * 17;
    const float m = (float)masks[p];
    const int cy = cyx[2 * b], cx = cyx[2 * b + 1];
    const int cpix = cy * WIDTH + cx;
    const float* labB = lab + (size_t)b * 86 * HW;
    const float grx = labB[(size_t)(18 + 2 * k) * HW + cpix];
    const float gry = labB[(size_t)(19 + 2 * k) * HW + cpix];
    const float* regB = regs + (size_t)b * 34 * HW;
    const float prx = regB[(size_t)(2 * k)     * HW + cpix];
    const float pry = regB[(size_t)(2 * k + 1) * HW + cpix];
    reg_num += m * (fabsf(prx - grx) + fabsf(pry - gry));
    msum += m;
    // kx/ky: round-half-even (rintf matches jnp.round), clip in float, then cast
    const float kxf = fminf(fmaxf(rintf(grx + (float)cx), 0.f), (float)(WIDTH - 1));
    const float kyf = fminf(fmaxf(rintf(gry + (float)cy), 0.f), (float)(WIDTH - 1));
    const int kpix = (int)kyf * WIDTH + (int)kxf;
    const float g0 = labB[(size_t)(52 + 2 * k) * HW + kpix];
    const float g1 = labB[(size_t)(53 + 2 * k) * HW + kpix];
    const float* offB = offs + (size_t)b * 34 * HW;
    const float o0 = offB[(size_t)(2 * k)     * HW + kpix];
    const float o1 = offB[(size_t)(2 * k + 1) * HW + kpix];
    off_num += m * (fabsf(o0 - g0) + fabsf(o1 - g1));
  }
  __shared__ float r0[256], r1[256], r2[256];
  r0[tid] = reg_num; r1[tid] = off_num; r2[tid] = msum; __syncthreads();
  for (int s = 128; s > 0; s >>= 1) {
    if (tid < s) { r0[tid] += r0[tid + s]; r1[tid] += r1[tid + s]; r2[tid] += r2[tid + s]; }
    __syncthreads();
  }
  if (tid == 0) { s3[0] = r0[0]; s3[1] = r1[0]; s3[2] = r2[0]; }
}

// ---------------- Kernel 4: final combine -------------------------------------------
__global__ __launch_bounds__(256) void k_final(const float* __restrict__ p_hm,
                                               const float* __restrict__ p_ct,
                                               const float* __restrict__ p_bone,
                                               const float* __restrict__ s3,
                                               float* __restrict__ out) {
  const int tid = threadIdx.x;
  float a = 0.f, b = 0.f, c = 0.f;
  for (int i = tid; i < NB2; i += 256) { a += p_hm[i]; b += p_ct[i]; c += p_bone[i]; }
  __shared__ float r0[256], r1[256], r2[256];
  r0[tid] = a; r1[tid] = b; r2[tid] = c; __syncthreads();
  for (int s = 128; s > 0; s >>= 1) {
    if (tid < s) { r0[tid] += r0[tid + s]; r1[tid] += r1[tid + s]; r2[tid] += r2[tid + s]; }
    __syncthreads();
  }
  if (tid == 0) {
    const float loss_hm   = r0[0] / 5013504.f;    // 128*17*48*48
    const float loss_ct   = r1[0] / 294912.f;     // 128*1*48*48
    const float loss_bone = r2[0] / 4718592.f;    // 128*16*48*48
    const float denom = 2.f * s3[2] + 1e-4f;      // sum(mask_reg)+eps == 2*sum(maskf)+eps
    out[0] = loss_hm + loss_ct + 1.5f * (s3[0] / denom) + (s3[1] / denom) + 0.15f * loss_bone;
  }
}

extern "C" void kernel_launch(void* const* d_in, const int* in_sizes, int n_in,
                              void* d_out, int out_size, void* d_ws, size_t ws_size,
                              hipStream_t stream) {
  const float* hm    = (const float*)d_in[0];
  const float* ctr   = (const float*)d_in[1];
  const float* regs  = (const float*)d_in[2];
  const float* offs  = (const float*)d_in[3];
  const float* lab   = (const float*)d_in[4];
  const int*   masks = (const int*)d_in[5];

  float* wsf    = (float*)d_ws;
  float* p_hm   = wsf;
  float* p_ct   = wsf + NB2;
  float* p_bone = wsf + 2 * NB2;
  float* s3     = wsf + 3 * NB2;          // 3 scalars (+1 pad)
  int*   cyx    = (int*)(wsf + 3 * NB2 + 4);  // 256 ints

  k_argmax    <<<NBATCH, 256, 0, stream>>>(lab, cyx);
  k_focal_bone<<<NB2,    256, 0, stream>>>(hm, ctr, lab, p_hm, p_ct, p_bone);
  k_regoff    <<<1,      256, 0, stream>>>(regs, offs, lab, masks, cyx, s3);
  k_final     <<<1,      256, 0, stream>>>(p_hm, p_ct, p_bone, s3, (float*)d_out);
}